// EwaldPotential_48661979463963
// MI455X (gfx1250) — compile-verified
//
#include <hip/hip_runtime.h>

#define NA 4096

typedef float v2f __attribute__((ext_vector_type(2)));
typedef float v8f __attribute__((ext_vector_type(8)));

__constant__ const float kInvSqrt2 = 0.70710678118654752f;   // 1/(sigma*sqrt(2)), sigma=1
__constant__ const float kEps      = 1e-6f;
__constant__ const float kInv2Pi   = 0.15915494309189535f;
__constant__ const float kInv4Pi   = 0.07957747154594767f;
__constant__ const float kSelfC    = 0.06349363593424097f;   // 1/(sigma*(2*pi)^1.5)
__constant__ const float kLog2e    = 1.4426950408889634f;
__constant__ const float kFltMax   = 3.402823466e38f;

// Abramowitz & Stegun 7.1.26 erf approximation, x >= 0.
__device__ __forceinline__ float erf_fast(float x) {
    float t = __builtin_amdgcn_rcpf(__builtin_fmaf(0.3275911f, x, 1.0f));
    float p = 1.061405429f;
    p = __builtin_fmaf(p, t, -1.453152027f);
    p = __builtin_fmaf(p, t,  1.421413741f);
    p = __builtin_fmaf(p, t, -0.284496736f);
    p = __builtin_fmaf(p, t,  0.254829592f);
    p *= t;
    float e = __builtin_amdgcn_exp2f(-x * x * kLog2e);   // exp(-x^2)
    return __builtin_fmaf(-p, e, 1.0f);
}

// Process one 16x16 d^2 tile: w = erf(d/sqrt2)/(d+eps), acc += q_i * w.
__device__ __forceinline__ float tile_accum(const v8f& d2, const float* __restrict__ q,
                                            int row_base, float acc) {
#pragma unroll
    for (int r = 0; r < 8; ++r) {
        float dd = __builtin_amdgcn_fmed3f(d2[r], 0.0f, kFltMax);  // single v_med3_f32 clamp
        float d  = __builtin_amdgcn_sqrtf(dd);
        float w  = erf_fast(d * kInvSqrt2) * __builtin_amdgcn_rcpf(d + kEps);
        if (dd < 1e-12f) w = 0.0f;                                 // exact zero on diagonal
        acc = __builtin_fmaf(q[row_base + r], w, acc);
    }
    return acc;
}

// Kernel 1: |p_i|^2 into ws, zero the column accumulator (ws is poisoned by harness).
__global__ void __launch_bounds__(256)
ewald_init(const float* __restrict__ pos, float* __restrict__ rsq, float* __restrict__ s) {
    int i = blockIdx.x * blockDim.x + threadIdx.x;
    if (i < NA) {
        float x = pos[3 * i + 0];
        float y = pos[3 * i + 1];
        float z = pos[3 * i + 2];
        rsq[i] = __builtin_fmaf(x, x, __builtin_fmaf(y, y, z * z));
        s[i] = 0.0f;
    }
}

// Kernel 2: tiled pairwise sum. One wave per 16-column j-tile per i-chunk.
// d^2 tile (16x16) from one V_WMMA_F32_16X16X4_F32:
//   A[m,:] = (x_i, y_i, z_i, 1)   B[:,n] = (-2x_j, -2y_j, -2z_j, |p_j|^2)   C[m,n] = |p_i|^2
// i-loop unrolled 2x: two WMMAs per iteration against the same resident B operand.
__global__ void __launch_bounds__(128)
ewald_tile(const float* __restrict__ pos, const float* __restrict__ q,
           const float* __restrict__ rsq, float* __restrict__ s) {
    const int lane = threadIdx.x & 31;
    const int wave = threadIdx.x >> 5;
    const int j0   = (blockIdx.x * 4 + wave) * 16;   // column tile base
    const int l15  = lane & 15;
    const int hi   = (lane >= 16) ? 1 : 0;           // upper half-wave
    const int moff = hi ? 8 : 0;                     // C/D row offset for upper half
    const int ka   = hi ? 2 : 0;                     // K-index base per half-wave

    // B operand (fixed per wave): VGPR0/1 lanes 0-15 hold K=0,1; lanes 16-31 K=2,3.
    v2f b;
    {
        const int j = j0 + l15;
        b.x = -2.0f * pos[3 * j + ka];
        b.y = hi ? rsq[j] : (-2.0f * pos[3 * j + 1]);
    }

    float acc = 0.0f;                                 // partial of s_{j0+l15} over my rows
    const int i_begin = blockIdx.y * (NA / 8);
    const int i_end   = i_begin + (NA / 8);

    for (int i0 = i_begin; i0 < i_end; i0 += 32) {
        const int i1 = i0 + 16;

        // A operands for the two 16-row blocks: lane<16 -> (x,y); lane>=16 -> (z,1)
        v2f a0, a1;
        {
            const int ia = i0 + l15;
            const int ib = i1 + l15;
            a0.x = pos[3 * ia + ka];
            a0.y = hi ? 1.0f : pos[3 * ia + 1];
            a1.x = pos[3 * ib + ka];
            a1.y = hi ? 1.0f : pos[3 * ib + 1];
        }
        // C operands: row-constant |p_i|^2 per the 16x16 f32 C/D layout.
        v8f c0, c1;
#pragma unroll
        for (int r = 0; r < 8; ++r) {
            c0[r] = rsq[i0 + moff + r];
            c1[r] = rsq[i1 + moff + r];
        }

        v8f d2a = __builtin_amdgcn_wmma_f32_16x16x4_f32(
            false, a0, false, b, (short)0, c0, false, false);
        v8f d2b = __builtin_amdgcn_wmma_f32_16x16x4_f32(
            false, a1, false, b, (short)0, c1, false, false);

        acc = tile_accum(d2a, q, i0 + moff, acc);
        acc = tile_accum(d2b, q, i1 + moff, acc);
    }

    // Combine the two half-wave partials for the same column n = l15 (wave32).
    acc += __shfl_xor(acc, 16, 32);
    if (lane < 16) atomicAdd(&s[j0 + l15], acc);
}

// Kernel 3: field + self terms; LDS tree reduction for the scalar potential.
__global__ void __launch_bounds__(512)
ewald_finalize(const float* __restrict__ q, const float* __restrict__ s,
               float* __restrict__ out) {
    __shared__ float red[512];
    const int t = threadIdx.x;
    float p = 0.0f;
    for (int j = t; j < NA; j += 512) {
        float sj = s[j];
        float qj = q[j];
        out[1 + j] = __builtin_fmaf(sj, kInv2Pi, qj * (2.0f * kSelfC));
        p = __builtin_fmaf(qj * sj, kInv4Pi, p);
        p = __builtin_fmaf(qj * qj, kSelfC, p);
    }
    red[t] = p;
    __syncthreads();
    for (int off = 256; off > 0; off >>= 1) {
        if (t < off) red[t] += red[t + off];
        __syncthreads();
    }
    if (t == 0) out[0] = red[0];
}

extern "C" void kernel_launch(void* const* d_in, const int* in_sizes, int n_in,
                              void* d_out, int out_size, void* d_ws, size_t ws_size,
                              hipStream_t stream) {
    const float* pos = (const float*)d_in[0];   // [N,3] f32
    const float* q   = (const float*)d_in[1];   // [N,1] f32
    float* out = (float*)d_out;                 // [N+1] f32
    float* rsq = (float*)d_ws;                  // N floats
    float* s   = rsq + NA;                      // N floats

    ewald_init<<<NA / 256, 256, 0, stream>>>(pos, rsq, s);

    dim3 grid(64, 8);                           // 64 blocks x 4 waves = 256 j-tiles; 8 i-chunks
    ewald_tile<<<grid, 128, 0, stream>>>(pos, q, rsq, s);

    ewald_finalize<<<1, 512, 0, stream>>>(q, s, out);
}